// MACE_58729382805670
// MI455X (gfx1250) — compile-verified
//
#include <hip/hip_runtime.h>
#include <hip/hip_bf16.h>

typedef __attribute__((ext_vector_type(2))) float v2f;
typedef __attribute__((ext_vector_type(4))) float v4f;
typedef __attribute__((ext_vector_type(8))) float v8f;

#define NN   16384
#define CC   128
#define SS   88
#define TT   336            // coefficient terms per (species, channel)
#define TG   84             // TT/4 packed term-groups
#define KTOT 16             // stacked mul dimension: 4+6+2+2+1+1

// workspace layout (bytes)
#define OFF_CTAB 0u                              // 88*336*128*4 = 15,138,816
#define OFF_AMAT 15138816u                       // 336*16*4     = 21,504
#define OFF_WSTK 15160320u                       // 88*16*128*4  = 720,896

// ---------------------------------------------------------------------------
// Build block-diagonal basis matrix Amat[336][16] from the six U tensors.
// Row t is ordered exactly in main-kernel consumption order:
//   0e block (t<84):   t = a*21 + {0:C1 | 1+b*5+{0:C2 | 1+j:C3}}
//   1o block (t>=84):  t = 84 + i*84 + same inner layout
// Column slots: k0..3 = w3_0e, k4..9 = w3_1o, k10..11 = w2_0e,
//               k12..13 = w2_1o, k14 = w1_0e, k15 = w1_1o
// ---------------------------------------------------------------------------
__global__ void mace_build_A(const float* __restrict__ u1_0e,
                             const float* __restrict__ u1_1o,
                             const float* __restrict__ u2_0e,
                             const float* __restrict__ u2_1o,
                             const float* __restrict__ u3_0e,
                             const float* __restrict__ u3_1o,
                             float* __restrict__ Amat) {
  int t = blockIdx.x * blockDim.x + threadIdx.x;
  if (t >= TT) return;
  float row[KTOT];
#pragma unroll
  for (int kk = 0; kk < KTOT; ++kk) row[kk] = 0.0f;

  int tt = t, ir = 0, i = 0;
  if (tt >= 84) { ir = 1; i = (tt - 84) / 84; tt = (tt - 84) % 84; }
  int a = tt / 21;
  int r = tt % 21;
  if (r == 0) {
    if (ir == 0) row[14] = u1_0e[a];            // (4,1,1)
    else         row[15] = u1_1o[a * 3 + i];    // (4,1,3)
  } else {
    int b = (r - 1) / 5;
    int q = (r - 1) % 5;
    if (q == 0) {
      if (ir == 0) {                            // u2_0e (4,4,2,1)
        row[10] = u2_0e[(a * 4 + b) * 2 + 0];
        row[11] = u2_0e[(a * 4 + b) * 2 + 1];
      } else {                                  // u2_1o (4,4,2,3)
        row[12] = u2_1o[((a * 4 + b) * 2 + 0) * 3 + i];
        row[13] = u2_1o[((a * 4 + b) * 2 + 1) * 3 + i];
      }
    } else {
      int j = q - 1;
      if (ir == 0) {                            // u3_0e (4,4,4,4,1)
#pragma unroll
        for (int k = 0; k < 4; ++k)
          row[k] = u3_0e[((a * 4 + b) * 4 + j) * 4 + k];
      } else {                                  // u3_1o (4,4,4,6,3)
#pragma unroll
        for (int k = 0; k < 6; ++k)
          row[4 + k] = u3_1o[(((a * 4 + b) * 4 + j) * 6 + k) * 3 + i];
      }
    }
  }
#pragma unroll
  for (int kk = 0; kk < KTOT; ++kk) Amat[t * KTOT + kk] = row[kk];
}

// ---------------------------------------------------------------------------
// Stack species weights into Wstk[s][16][128]
// ---------------------------------------------------------------------------
__global__ void mace_build_W(const float* __restrict__ w1_0e,
                             const float* __restrict__ w1_1o,
                             const float* __restrict__ w2_0e,
                             const float* __restrict__ w2_1o,
                             const float* __restrict__ w3_0e,
                             const float* __restrict__ w3_1o,
                             float* __restrict__ Wstk) {
  int e = blockIdx.x * blockDim.x + threadIdx.x;   // 88*16*128 = 180224
  if (e >= SS * KTOT * CC) return;
  int s  = e >> 11;          // /(16*128)
  int kk = (e >> 7) & 15;
  int c  = e & 127;
  float v;
  if      (kk < 4)   v = w3_0e[(s * 4 + kk)        * CC + c];
  else if (kk < 10)  v = w3_1o[(s * 6 + (kk - 4))  * CC + c];
  else if (kk < 12)  v = w2_0e[(s * 2 + (kk - 10)) * CC + c];
  else if (kk < 14)  v = w2_1o[(s * 2 + (kk - 12)) * CC + c];
  else if (kk == 14) v = w1_0e[s * CC + c];
  else               v = w1_1o[s * CC + c];
  Wstk[e] = v;
}

// ---------------------------------------------------------------------------
// Ctab[s] = Amat[336x16] * Wstk[s][16x128] via v_wmma_f32_16x16x4_f32.
// One wave per 16x16 output tile; 21x8 tiles per species, 88 species
// -> 14784 waves -> 1848 blocks of 8 waves (exact, no guard; EXEC all ones).
// Output stored packed-by-4 terms: Ctab[s][t>>2][c][t&3].
// ---------------------------------------------------------------------------
__global__ void mace_coeff_gemm(const float* __restrict__ Amat,
                                const float* __restrict__ Wstk,
                                float* __restrict__ Ctab) {
  int wave = threadIdx.x >> 5;
  int lane = threadIdx.x & 31;
  int gw   = blockIdx.x * 8 + wave;          // 0 .. 14783
  int s    = gw / 168;
  int tIdx = gw % 168;
  int mt   = tIdx >> 3;                      // 0..20
  int nt   = tIdx & 7;                       // 0..7
  int hi   = lane >> 4;                      // 0/1
  int lo   = lane & 15;
  int row  = mt * 16 + lo;                   // A-matrix row for this lane
  int col  = nt * 16 + lo;                   // B / D column for this lane

  v8f acc = {};
#pragma unroll
  for (int ks = 0; ks < 4; ++ks) {
    int kA = ks * 4 + 2 * hi;
    v2f av, bv;
    av.x = Amat[row * KTOT + kA];
    av.y = Amat[row * KTOT + kA + 1];
    bv.x = Wstk[(s * KTOT + kA) * CC + col];
    bv.y = Wstk[(s * KTOT + kA + 1) * CC + col];
    acc = __builtin_amdgcn_wmma_f32_16x16x4_f32(
        /*neg_a=*/false, av, /*neg_b=*/false, bv,
        /*c_mod=*/(short)0, acc, /*reuse_a=*/false, /*reuse_b=*/false);
  }
#pragma unroll
  for (int v = 0; v < 8; ++v) {
    int tr = mt * 16 + v + 8 * hi;           // D layout: M = v (+8 for hi lanes)
    int g  = tr >> 2;                        // packed term-group
    int sl = tr & 3;
    Ctab[(((size_t)s * TG + g) * CC + col) * 4 + sl] = acc[v];
  }
}

// ---------------------------------------------------------------------------
// Main pass: one thread per (atom, channel). Lanes span consecutive channels.
// Coefficients consumed strictly sequentially -> 84 coalesced b128 loads per
// thread (each wave-load = contiguous 512B from the L2-resident table).
// x / out use non-temporal hints to keep Ctab resident in L2.
// ---------------------------------------------------------------------------
__global__ void __launch_bounds__(256)
mace_main(const float* __restrict__ x,
          const int*   __restrict__ species,
          const float* __restrict__ Ctab,
          float* __restrict__ out) {
  int gid = blockIdx.x * blockDim.x + threadIdx.x;  // 0 .. N*C-1
  int n = gid >> 7;
  int c = gid & 127;
  int s = species[n];

  const v4f xv = __builtin_nontemporal_load((const v4f*)x + gid);
  float xj[4] = {xv.x, xv.y, xv.z, xv.w};

  // packed coefficient stream: group g lives at p4[g*CC]
  const v4f* p4 = (const v4f*)Ctab + ((size_t)s * TG * CC + c);
  v4f cur;
  int tcnt = 0;
  // NEXT(): returns coefficient t=tcnt, loading one float4 per 4 terms.
  // Fully unrolled below, so slot indices and offsets constant-fold.
#define NEXT(dst)                                   \
  {                                                 \
    int _sl = tcnt & 3;                             \
    if (_sl == 0) { cur = p4[0]; p4 += CC; }        \
    dst = cur[_sl];                                 \
    ++tcnt;                                         \
  }

  // ---- 0e ----
  float acc0 = 0.0f;
#pragma unroll
  for (int a = 0; a < 4; ++a) {
    float Ba; NEXT(Ba);                             // C1 term
#pragma unroll
    for (int b = 0; b < 4; ++b) {
      float Aab; NEXT(Aab);                         // C2 term
#pragma unroll
      for (int j = 0; j < 4; ++j) {                 // C3 terms
        float c3; NEXT(c3);
        Aab = __builtin_fmaf(c3, xj[j], Aab);
      }
      Ba = __builtin_fmaf(Aab, xj[b], Ba);
    }
    acc0 = __builtin_fmaf(Ba, xj[a], acc0);
  }
  __builtin_nontemporal_store(acc0, out + (size_t)n * 512 + c);

  // ---- 1o (i = 0..2) ----
  float o1[3];
#pragma unroll
  for (int i = 0; i < 3; ++i) {
    float acc1 = 0.0f;
#pragma unroll
    for (int a = 0; a < 4; ++a) {
      float Ba; NEXT(Ba);
#pragma unroll
      for (int b = 0; b < 4; ++b) {
        float Aab; NEXT(Aab);
#pragma unroll
        for (int j = 0; j < 4; ++j) {
          float c3; NEXT(c3);
          Aab = __builtin_fmaf(c3, xj[j], Aab);
        }
        Ba = __builtin_fmaf(Aab, xj[b], Ba);
      }
      acc1 = __builtin_fmaf(Ba, xj[a], acc1);
    }
    o1[i] = acc1;
  }
#undef NEXT
  float* o = out + (size_t)n * 512 + 128 + c * 3;
  __builtin_nontemporal_store(o1[0], o + 0);
  __builtin_nontemporal_store(o1[1], o + 1);
  __builtin_nontemporal_store(o1[2], o + 2);
}

// ---------------------------------------------------------------------------
extern "C" void kernel_launch(void* const* d_in, const int* in_sizes, int n_in,
                              void* d_out, int out_size, void* d_ws, size_t ws_size,
                              hipStream_t stream) {
  // setup_inputs() dict order:
  // 0:x 1:u1_0e 2:w1_0e 3:u1_1o 4:w1_1o 5:u2_0e 6:w2_0e 7:u2_1o 8:w2_1o
  // 9:u3_0e 10:w3_0e 11:u3_1o 12:w3_1o 13:species
  const float* x      = (const float*)d_in[0];
  const float* u1_0e  = (const float*)d_in[1];
  const float* w1_0e  = (const float*)d_in[2];
  const float* u1_1o  = (const float*)d_in[3];
  const float* w1_1o  = (const float*)d_in[4];
  const float* u2_0e  = (const float*)d_in[5];
  const float* w2_0e  = (const float*)d_in[6];
  const float* u2_1o  = (const float*)d_in[7];
  const float* w2_1o  = (const float*)d_in[8];
  const float* u3_0e  = (const float*)d_in[9];
  const float* w3_0e  = (const float*)d_in[10];
  const float* u3_1o  = (const float*)d_in[11];
  const float* w3_1o  = (const float*)d_in[12];
  const int*   spec   = (const int*)d_in[13];
  float* out = (float*)d_out;

  char* ws = (char*)d_ws;
  float* Ctab = (float*)(ws + OFF_CTAB);
  float* Amat = (float*)(ws + OFF_AMAT);
  float* Wstk = (float*)(ws + OFF_WSTK);

  // 1) basis matrix [336 x 16]
  mace_build_A<<<2, 256, 0, stream>>>(u1_0e, u1_1o, u2_0e, u2_1o, u3_0e, u3_1o, Amat);
  // 2) stacked weights [88 x 16 x 128]
  mace_build_W<<<(SS * KTOT * CC + 255) / 256, 256, 0, stream>>>(
      w1_0e, w1_1o, w2_0e, w2_1o, w3_0e, w3_1o, Wstk);
  // 3) per-species coefficient tables via WMMA: 14784 tiles / 8 waves per block
  mace_coeff_gemm<<<1848, 256, 0, stream>>>(Amat, Wstk, Ctab);
  // 4) main channelwise polynomial pass
  mace_main<<<(NN * CC) / 256, 256, 0, stream>>>(x, spec, Ctab, out);
}